// MyModel_87522843561235
// MI455X (gfx1250) — compile-verified
//
#include <hip/hip_runtime.h>

typedef __attribute__((ext_vector_type(2))) float v2f;
typedef __attribute__((ext_vector_type(8))) float v8f;
typedef __attribute__((ext_vector_type(2))) int   v2i;

#define KD 64          // state dimension
#define NTHREADS 512   // 16 wave32 waves -> 4x4 grid of 16x16 WMMA tiles

#if __has_builtin(__builtin_amdgcn_global_load_async_to_lds_b64) && \
    __has_builtin(__builtin_amdgcn_s_wait_asynccnt)
#define HAVE_ASYNC_LDS 1
typedef v2i __attribute__((address_space(1)))* g64_ptr;   // global b64 element
typedef v2i __attribute__((address_space(3)))* l64_ptr;   // LDS b64 element
#else
#define HAVE_ASYNC_LDS 0
#endif

__launch_bounds__(NTHREADS, 1)
__global__ void kalman_scan_wmma(const float* __restrict__ emissions, // [N]
                                 const float* __restrict__ Ht,        // [N,64]
                                 const float* __restrict__ Fg,        // [64,64]
                                 const float* __restrict__ Qg,        // [64,64]
                                 const float* __restrict__ Rg,        // [1]
                                 const float* __restrict__ mu0,       // [64]
                                 const float* __restrict__ Sigma0,    // [64,64]
                                 float* __restrict__ out,             // [N,64]
                                 int N)
{
    __shared__ float Psh[KD * KD];   // P, then M = F P F^T + Q, then P_next (in place)
    __shared__ float Tsh[KD * KD];   // T = P * F^T
    __shared__ float Fsh[KD * KD];   // transition matrix (kept for F@mu partials)
    __shared__ float pw[4][KD];      // partials of w = H P
    __shared__ float pf[4][KD];      // partials of F @ mu
    __shared__ float pg[4][KD];      // partials of g = T^T h (= F w)
    __shared__ float hsh[KD];        // H_t row
    __shared__ float wsh[KD];        // w = H P
    __shared__ float swh[KD];        // w[j]*h[j] (for S reduction)
    __shared__ float hmn[KD];        // h[j]*mu[j] (for innovation reduction)
    __shared__ float gsh[KD];        // g = F w
    __shared__ float fmush[KD];      // F @ mu
    __shared__ float mush[KD];       // current mean
    __shared__ float sc[8];          // [0]=1/S [1]=R [3]=h.mu [4]/[5]=y double-buffered

    const int tid  = threadIdx.x;
    const int lane = tid & 31;
    const int wave = tid >> 5;       // 0..15
    const int hi   = lane >> 4;      // lane half
    const int ln   = lane & 15;
    const int rowB = (wave >> 2) * 16;
    const int colB = (wave & 3) * 16;

    // ---- one-time init ----
    float qreg[8];
#pragma unroll
    for (int r = 0; r < 8; ++r)
        qreg[r] = Qg[(rowB + r + 8 * hi) * KD + colB + ln];

    for (int i = tid; i < KD * KD; i += NTHREADS) {
        Fsh[i] = Fg[i];
        Psh[i] = Sigma0[i];
    }
    if (tid < KD) { mush[tid] = mu0[tid]; hsh[tid] = Ht[tid]; }
    if (tid == 0) { sc[1] = Rg[0]; sc[4] = emissions[0]; }
    __syncthreads();

    // ---- F is constant: cache both WMMA operand layouts in registers ----
    v2f faT[16];  // phase A B-operand: B[k][n] = F[n][k]  (F^T tile)
    v2f fa[16];   // phase B A-operand: F tile rows
#pragma unroll
    for (int kb = 0; kb < 16; ++kb) {
        const int k0 = kb * 4 + 2 * hi;
        faT[kb].x = Fsh[(colB + ln) * KD + k0];
        faT[kb].y = Fsh[(colB + ln) * KD + k0 + 1];
        fa[kb].x  = Fsh[(rowB + ln) * KD + k0];
        fa[kb].y  = Fsh[(rowB + ln) * KD + k0 + 1];
    }

    for (int t = 0; t < N; ++t) {
        // ======== Phase A: WMMA  T = P @ F^T ; partials pf=F@mu, pw=H P ========
        {
            const int aRow = (rowB + ln) * KD;
            v8f acc0 = {}, acc1 = {};
#pragma unroll
            for (int kb = 0; kb < 16; kb += 2) {   // two accumulators: halve RAW chain
                const int k0 = kb * 4 + 2 * hi;
                v2f a;
                a.x = Psh[aRow + k0];     a.y = Psh[aRow + k0 + 1];
                acc0 = __builtin_amdgcn_wmma_f32_16x16x4_f32(false, a, false, faT[kb],
                                                             (short)0, acc0, false, false);
                const int k1 = k0 + 4;
                a.x = Psh[aRow + k1];     a.y = Psh[aRow + k1 + 1];
                acc1 = __builtin_amdgcn_wmma_f32_16x16x4_f32(false, a, false, faT[kb + 1],
                                                             (short)0, acc1, false, false);
            }
#pragma unroll
            for (int r = 0; r < 8; ++r)
                Tsh[(rowB + r + 8 * hi) * KD + colB + ln] = acc0[r] + acc1[r];
        }
        {   // vector partials in the shadow of the matmul (16-serial each)
            const int idx = tid & 63;
            const int j0  = ((tid >> 6) & 3) * 16;
            if (tid < 256) {                       // pf[p][i] = sum_{j in p} F[i][j]*mu[j]
                float a = 0.f;
#pragma unroll
                for (int j = 0; j < 16; ++j) a += Fsh[idx * KD + j0 + j] * mush[j0 + j];
                pf[(tid >> 6) & 3][idx] = a;
            } else {                               // pw[p][j] = sum_{i in p} h[i]*P[i][j]
                float a = 0.f;
#pragma unroll
                for (int i = 0; i < 16; ++i) a += hsh[j0 + i] * Psh[(j0 + i) * KD + idx];
                pw[(tid >> 6) & 3][idx] = a;
            }
        }
        __syncthreads();

        // ======== Phase B: WMMA  M = F @ T + Q -> Psh ; w/hm/pg/fmu finals ========
        {
            v8f acc0, acc1 = {};
#pragma unroll
            for (int r = 0; r < 8; ++r) acc0[r] = qreg[r];   // C = Q tile
#pragma unroll
            for (int kb = 0; kb < 16; kb += 2) {
                const int k0 = kb * 4 + 2 * hi;
                v2f b;
                b.x = Tsh[k0 * KD + colB + ln]; b.y = Tsh[(k0 + 1) * KD + colB + ln];
                acc0 = __builtin_amdgcn_wmma_f32_16x16x4_f32(false, fa[kb], false, b,
                                                             (short)0, acc0, false, false);
                const int k1 = k0 + 4;
                b.x = Tsh[k1 * KD + colB + ln]; b.y = Tsh[(k1 + 1) * KD + colB + ln];
                acc1 = __builtin_amdgcn_wmma_f32_16x16x4_f32(false, fa[kb + 1], false, b,
                                                             (short)0, acc1, false, false);
            }
#pragma unroll
            for (int r = 0; r < 8; ++r)
                Psh[(rowB + r + 8 * hi) * KD + colB + ln] = acc0[r] + acc1[r];
        }
        if (tid < 64) {                            // w final + S terms
            const float w = pw[0][tid] + pw[1][tid] + pw[2][tid] + pw[3][tid];
            wsh[tid] = w;
            swh[tid] = w * hsh[tid];
        } else if (tid < 128) {                    // innovation terms
            const int j = tid - 64;
            hmn[j] = hsh[j] * mush[j];
        } else if (tid < 384) {                    // pg[p][i] = sum_{j in p} T[j][i]*h[j]
            const int t2 = tid - 128, i = t2 & 63, j0 = (t2 >> 6) * 16;
            float a = 0.f;
#pragma unroll
            for (int j = 0; j < 16; ++j) a += Tsh[(j0 + j) * KD + i] * hsh[j0 + j];
            pg[t2 >> 6][i] = a;
        } else if (tid < 448) {                    // fmu final
            const int i = tid - 384;
            fmush[i] = pf[0][i] + pf[1][i] + pf[2][i] + pf[3][i];
        }
        __syncthreads();

        // ======== Phase C: tree reductions (S, h.mu) + g final ========
        if (tid < 32) {                            // S = R + jitter + sum w[j]h[j]
            float v = swh[lane] + swh[lane + 32];
            v += __shfl_xor(v, 16, 32); v += __shfl_xor(v, 8, 32);
            v += __shfl_xor(v, 4, 32);  v += __shfl_xor(v, 2, 32);
            v += __shfl_xor(v, 1, 32);
            if (lane == 0) sc[0] = 1.f / (sc[1] + 1e-6f + v);
        } else if (tid < 64) {                     // h . mu
            float v = hmn[lane] + hmn[lane + 32];
            v += __shfl_xor(v, 16, 32); v += __shfl_xor(v, 8, 32);
            v += __shfl_xor(v, 4, 32);  v += __shfl_xor(v, 2, 32);
            v += __shfl_xor(v, 1, 32);
            if (lane == 0) sc[3] = v;
        } else if (tid < 128) {                    // g final
            const int i = tid - 64;
            gsh[i] = pg[0][i] + pg[1][i] + pg[2][i] + pg[3][i];
        }
        __syncthreads();

        // ======== Phase D: P -= g g^T / S ; mean update + output ; next h,y ========
        {
#if HAVE_ASYNC_LDS
            // Issue async LDS fill of next H row first, overlap with downdate work.
            if (t + 1 < N && wave == 14) {
                g64_ptr src = (g64_ptr)(unsigned long long)
                              (Ht + (size_t)(t + 1) * KD + 2 * lane);
                l64_ptr dst = (l64_ptr)(unsigned)(unsigned long long)(&hsh[2 * lane]);
                __builtin_amdgcn_global_load_async_to_lds_b64(src, dst, 0, 0);
            }
#endif
            const float invS = sc[0];
            const float y    = sc[4 + (t & 1)];
            const float kfac = (y - sc[3]) * invS;
            const int row  = tid >> 3;
            const int col0 = (tid & 7) << 3;
            const float gi = gsh[row] * invS;
#pragma unroll
            for (int c = 0; c < 8; ++c)
                Psh[row * KD + col0 + c] -= gi * gsh[col0 + c];
            if (tid < 64) {
                out[(size_t)t * KD + tid] = mush[tid] + kfac * wsh[tid]; // mu_c (filtered)
                mush[tid] = fmush[tid] + kfac * gsh[tid];                // mu_p = F mu_c
            }
            if (t + 1 < N) {
#if !HAVE_ASYNC_LDS
                if (tid >= 448) hsh[tid - 448] = Ht[(size_t)(t + 1) * KD + (tid - 448)];
#endif
                if (tid == 447) sc[4 + ((t + 1) & 1)] = emissions[t + 1];
                if (tid == 446 && t + 2 < N)
                    __builtin_prefetch(&Ht[(size_t)(t + 2) * KD], 0, 3);
            }
#if HAVE_ASYNC_LDS
            if (wave == 14) __builtin_amdgcn_s_wait_asynccnt(0); // publish hsh before barrier
#endif
        }
        __syncthreads();
    }
}

extern "C" void kernel_launch(void* const* d_in, const int* in_sizes, int n_in,
                              void* d_out, int out_size, void* d_ws, size_t ws_size,
                              hipStream_t stream) {
    const float* emissions = (const float*)d_in[0]; // [N]
    const float* Ht        = (const float*)d_in[1]; // [N,1,64]
    const float* F         = (const float*)d_in[2]; // [64,64]
    const float* Q         = (const float*)d_in[3]; // [64,64]
    const float* R         = (const float*)d_in[4]; // [1]
    const float* mu0       = (const float*)d_in[5]; // [64]
    const float* Sigma0    = (const float*)d_in[6]; // [64,64]
    float* out = (float*)d_out;
    const int N = in_sizes[0];

    // Sequential scan: one workgroup (16 wave32 waves) owns the whole chain.
    kalman_scan_wmma<<<1, NTHREADS, 0, stream>>>(emissions, Ht, F, Q, R,
                                                 mu0, Sigma0, out, N);
}